// Pool_41480794144814
// MI455X (gfx1250) — compile-verified
//
#include <hip/hip_runtime.h>
#include <cstdint>

typedef __attribute__((ext_vector_type(2))) float v2f;
typedef __attribute__((ext_vector_type(4))) float v4f;
typedef __attribute__((ext_vector_type(8))) float v8f;

// jax.random.key(42) -> threefry key (0, 42)
#define TF_KS0 0u
#define TF_KS1 42u
#define TF_KS2 (0x1BD11BDAu ^ TF_KS0 ^ TF_KS1)

__device__ __forceinline__ uint32_t rotl32(uint32_t x, uint32_t d) {
  return (x << d) | (x >> (32u - d));
}

// Threefry-2x32, 20 rounds — bit-exact with JAX's threefry2x32 block
__device__ __forceinline__ void threefry2x32(uint32_t c0, uint32_t c1,
                                             uint32_t& o0, uint32_t& o1) {
  uint32_t x0 = c0 + TF_KS0;
  uint32_t x1 = c1 + TF_KS1;
#define TF_ROUND(d) { x0 += x1; x1 = rotl32(x1, d); x1 ^= x0; }
  TF_ROUND(13) TF_ROUND(15) TF_ROUND(26) TF_ROUND(6)
  x0 += TF_KS1; x1 += TF_KS2 + 1u;
  TF_ROUND(17) TF_ROUND(29) TF_ROUND(16) TF_ROUND(24)
  x0 += TF_KS2; x1 += TF_KS0 + 2u;
  TF_ROUND(13) TF_ROUND(15) TF_ROUND(26) TF_ROUND(6)
  x0 += TF_KS0; x1 += TF_KS1 + 3u;
  TF_ROUND(17) TF_ROUND(29) TF_ROUND(16) TF_ROUND(24)
  x0 += TF_KS1; x1 += TF_KS2 + 4u;
  TF_ROUND(13) TF_ROUND(15) TF_ROUND(26) TF_ROUND(6)
  x0 += TF_KS2; x1 += TF_KS0 + 5u;
#undef TF_ROUND
  o0 = x0; o1 = x1;
}

// gfx1250 hardware V_TANH_F32 when the builtin exists; branch-free fallback
__device__ __forceinline__ float fast_tanh(float x) {
#if __has_builtin(__builtin_amdgcn_tanhf)
  return __builtin_amdgcn_tanhf(x);
#else
  // tanh(x) = sign(x) * (1 - 2/(exp(2|x|)+1)), branch-free via v_exp_f32
  float ax = __builtin_fabsf(x);
  float e  = __expf(2.0f * ax);           // v_exp_f32 path
  float t  = 1.0f - 2.0f / (e + 1.0f);
  return __builtin_copysignf(t, x);
#endif
}

// jax.random.gumbel: u = uniform(tiny, 1) from 23 mantissa bits; g = -log(-log(u))
__device__ __forceinline__ float gumbel_from_bits(uint32_t bits) {
  const float tiny = 1.17549435e-38f;
  float f = __uint_as_float(0x3f800000u | (bits >> 9)) - 1.0f; // [0,1)
  float u = f * (1.0f - tiny) + tiny;
  u = fmaxf(u, tiny);
  return -__logf(-__logf(u));             // v_log_f32 path
}

// Grid: 2^24 tiles total. Each wave handles two partner 16-tile strips:
//   strip0 = tiles [tb, tb+16)           (element indices < 2^25)
//   strip1 = tiles [tb+2^23, tb+2^23+16) (element indices = strip0 + 2^25)
// so each threefry block (c, c+2^25) serves one element of each strip.
__global__ __launch_bounds__(256) void
pool_wmma_kernel(const float* __restrict__ c,
                 const float* __restrict__ W1,
                 const float* __restrict__ W2,
                 float* __restrict__ pooled,
                 float* __restrict__ eta)
{
  const uint32_t lane   = threadIdx.x & 31u;
  const bool     lo     = lane < 16u;               // low half-wave flag
  const uint32_t waveId = blockIdx.x * 8u + (threadIdx.x >> 5); // < 2^19
  const uint32_t tb     = waveId * 16u;             // strip0 tile base, < 2^23

  // ---- weights in WMMA A-layout: rows 0-3 = W[q][p], rows 4-15 = 0 ----
  // A 16x4 f32: lane L<16: V0=K0,V1=K1 of row L; lane L+16: V0=K2,V1=K3.
  const uint32_t r  = lane & 15u;
  const uint32_t kk = (lane >> 4) << 1;             // 0 or 2
  v2f A1 = {}, A2 = {};
  if (r < 4u) {
    A1[0] = W1[r * 4u + kk]; A1[1] = W1[r * 4u + kk + 1u];
    A2[0] = W2[r * 4u + kk]; A2[1] = W2[r * 4u + kk + 1u];
  }

  // ---- strip loads: coalesced float2, lands directly in WMMA B-layout ----
  // B 4x16 f32: lane t: V0=p0,V1=p1 (tile t top row); lane t+16: V0=p2,V1=p3.
  const uint32_t plane = tb >> 14;                  // 16384 tiles / 256x256 plane
  const uint32_t trow  = (tb >> 7) & 127u;
  const uint32_t tcol  = tb & 127u;                 // multiple of 16
  const uint32_t prow  = 2u * trow + (lane >> 4);
  const uint32_t pcol  = 2u * (tcol + (lane & 15u));
  const size_t   off0  = ((size_t)plane << 16) + ((size_t)prow << 8) + pcol;
  const size_t   off1  = off0 + ((size_t)512u << 16);  // +2^25 elements

  v2f B0 = *(const v2f*)(c + off0);
  v2f B1 = *(const v2f*)(c + off1);

  const v8f zero8 = {};
  // ---- layer 1: D[q][t] = sum_p W1[q][p] * tile_t[p] (per-lane result!) ----
  v8f H0 = __builtin_amdgcn_wmma_f32_16x16x4_f32(false, A1, false, B0, (short)0, zero8, false, false);
  v8f H1 = __builtin_amdgcn_wmma_f32_16x16x4_f32(false, A1, false, B1, (short)0, zero8, false, false);

  float h00 = fast_tanh(H0[0]), h01 = fast_tanh(H0[1]);
  float h02 = fast_tanh(H0[2]), h03 = fast_tanh(H0[3]);
  float h10 = fast_tanh(H1[0]), h11 = fast_tanh(H1[1]);
  float h12 = fast_tanh(H1[2]), h13 = fast_tanh(H1[3]);

  // rebuild B-layout for layer 2: hi lanes need h2/h3 from their low partner
  float s02 = __shfl_xor(h02, 16, 32);
  float s03 = __shfl_xor(h03, 16, 32);
  float s12 = __shfl_xor(h12, 16, 32);
  float s13 = __shfl_xor(h13, 16, 32);
  v2f HB0, HB1;
  HB0[0] = lo ? h00 : s02;  HB0[1] = lo ? h01 : s03;
  HB1[0] = lo ? h10 : s12;  HB1[1] = lo ? h11 : s13;

  v8f L0 = __builtin_amdgcn_wmma_f32_16x16x4_f32(false, A2, false, HB0, (short)0, zero8, false, false);
  v8f L1 = __builtin_amdgcn_wmma_f32_16x16x4_f32(false, A2, false, HB1, (short)0, zero8, false, false);

  // ---- merge both strips onto all 32 lanes (lane t: strip0 tile t, lane t+16: strip1 tile t)
  float q0s = __shfl_xor(L1[0], 16, 32);
  float q1s = __shfl_xor(L1[1], 16, 32);
  float q2s = __shfl_xor(L1[2], 16, 32);
  float q3s = __shfl_xor(L1[3], 16, 32);
  float lg0 = lo ? L0[0] : q0s;
  float lg1 = lo ? L0[1] : q1s;
  float lg2 = lo ? L0[2] : q2s;
  float lg3 = lo ? L0[3] : q3s;

  // raw tile values for the pooled gather
  float b00s = __shfl_xor(B0[0], 16, 32);
  float b01s = __shfl_xor(B0[1], 16, 32);
  float b10s = __shfl_xor(B1[0], 16, 32);
  float b11s = __shfl_xor(B1[1], 16, 32);
  float tv0 = lo ? B0[0] : b10s;
  float tv1 = lo ? B0[1] : b11s;
  float tv2 = lo ? b00s  : B1[0];
  float tv3 = lo ? b01s  : B1[1];

  // ---- softmax (jax.nn.softmax: subtract max) ----
  float mx = fmaxf(fmaxf(lg0, lg1), fmaxf(lg2, lg3));
  float e0 = __expf(lg0 - mx), e1 = __expf(lg1 - mx);
  float e2 = __expf(lg2 - mx), e3 = __expf(lg3 - mx);
  float inv = 1.0f / (e0 + e1 + e2 + e3);

  // ---- threefry: pair (lane t, lane t+16) shares blocks (c, c+2^25) ----
  // bits[j] = x0 of block(j, j+2^25) for j < 2^25, else x1 of block(j-2^25, j).
  const uint32_t elo = (tb + (lane & 15u)) * 4u;  // low-half element base of pair
  const uint32_t cA  = elo + kk;                  // lo lane: q=0,1; hi lane: q=2,3
  uint32_t y0a, y1a, y0b, y1b;
  threefry2x32(cA,      cA      + 0x2000000u, y0a, y1a);
  threefry2x32(cA + 1u, cA + 1u + 0x2000000u, y0b, y1b);
  uint32_t sendA = lo ? y1a : y0a;   // lo sends its x1 (hi needs it), hi sends x0
  uint32_t sendB = lo ? y1b : y0b;
  uint32_t othA = __shfl_xor(sendA, 16, 32);
  uint32_t othB = __shfl_xor(sendB, 16, 32);
  uint32_t r0 = lo ? y0a : othA;
  uint32_t r1 = lo ? y0b : othB;
  uint32_t r2 = lo ? othA : y1a;
  uint32_t r3 = lo ? othB : y1b;

  // ---- gumbel-max categorical (first max wins, as jnp.argmax) ----
  float v0 = lg0 + gumbel_from_bits(r0);
  float v1 = lg1 + gumbel_from_bits(r1);
  float v2 = lg2 + gumbel_from_bits(r2);
  float v3 = lg3 + gumbel_from_bits(r3);
  int z = 0; float best = v0;
  if (v1 > best) { best = v1; z = 1; }
  if (v2 > best) { best = v2; z = 2; }
  if (v3 > best) { best = v3; z = 3; }
  float pv = (z == 0) ? tv0 : (z == 1) ? tv1 : (z == 2) ? tv2 : tv3;

  // ---- stores: pooled 4B/lane contiguous, eta float4/lane contiguous ----
  const uint32_t tile = lo ? (tb + lane) : (tb + (lane & 15u) + (1u << 23));
  pooled[tile] = pv;
  v4f ev; ev[0] = e0 * inv; ev[1] = e1 * inv; ev[2] = e2 * inv; ev[3] = e3 * inv;
  *(v4f*)(eta + ((size_t)tile << 2)) = ev;
}

extern "C" void kernel_launch(void* const* d_in, const int* in_sizes, int n_in,
                              void* d_out, int out_size, void* d_ws, size_t ws_size,
                              hipStream_t stream) {
  const float* c  = (const float*)d_in[0];   // [16,64,256,256] f32
  const float* W1 = (const float*)d_in[1];   // [4,4] f32
  const float* W2 = (const float*)d_in[2];   // [4,4] f32
  float* pooled = (float*)d_out;                        // 16*64*128*128 floats
  float* eta    = pooled + (size_t)16 * 64 * 128 * 128; // then 4x that
  // 2^19 waves; each wave covers 32 tiles (two partner 16-tile strips)
  dim3 grid(65536), block(256);
  pool_wmma_kernel<<<grid, block, 0, stream>>>(c, W1, W2, pooled, eta);
}